// NeighborStatOP_50371376447658
// MI455X (gfx1250) — compile-verified
//
#include <hip/hip_runtime.h>

typedef __attribute__((ext_vector_type(2))) float v2f;
typedef __attribute__((ext_vector_type(8))) float v8f;

#define NTYPES_K 4
#define RCUT2 36.0f

__global__ void ns_init_ws(int* wsMax, int n) {
    int i = blockIdx.x * blockDim.x + threadIdx.x;
    if (i < n) wsMax[i] = 0;
}

__global__ void ns_finalize(const int* __restrict__ wsMax, float* __restrict__ out, int n) {
    int i = blockIdx.x * blockDim.x + threadIdx.x;
    if (i < n) out[i] = (float)wsMax[i];
}

// One wave (32 lanes) owns one 16-row i-tile. The whole frame's precomputed
// (-2x,-2y,-2z,|c|^2) table + atom types are staged into LDS once per block.
// Each j-tile of 16 atoms is one V_WMMA_F32_16X16X4_F32:
//   A[m][k] = (x_i, y_i, z_i, 1)              (16x4)
//   B[k][n] = (-2x_j, -2y_j, -2z_j, |c_j|^2)  (4x16)
//   C[m][n] = |c_i|^2                         (accumulator)
//   D[m][n] = |c_i - c_j|^2
__global__ __launch_bounds__(64) void ns_main(const float* __restrict__ coord,
                                              const int* __restrict__ atype,
                                              float* __restrict__ minOut,
                                              int* __restrict__ wsMax,
                                              int nloc)
{
    extern __shared__ char smem[];
    float4* sC  = reinterpret_cast<float4*>(smem);                       // nloc float4
    float2* sC2 = reinterpret_cast<float2*>(smem);                       // pair view
    int*    sT  = reinterpret_cast<int*>(smem + (size_t)nloc * sizeof(float4));

    const int frame = blockIdx.y;
    const float* c = coord + (size_t)frame * nloc * 3;
    const int*   ty = atype + (size_t)frame * nloc;

    // Stage: -2*coord and squared norm, plus type.
    for (int a = threadIdx.x; a < nloc; a += blockDim.x) {
        float x = c[3 * a + 0], y = c[3 * a + 1], z = c[3 * a + 2];
        sC[a] = make_float4(-2.0f * x, -2.0f * y, -2.0f * z, x * x + y * y + z * z);
        sT[a] = ty[a];
    }
    __syncthreads();

    const int lane = threadIdx.x & 31;
    const int half = lane >> 4;   // lanes 0-15 vs 16-31
    const int col  = lane & 15;

    // Wave index is uniform within the wave: pin it to an SGPR so all
    // iTile-dependent control flow is scalar (s_cbranch, no exec masking).
    const int waveS = __builtin_amdgcn_readfirstlane(threadIdx.x >> 5);
    const int iTile = blockIdx.x * 2 + waveS;   // 2 waves per block
    const int i0 = iTile << 4;

    // A operand: lane holds row M=col, K-pair {0,1} (half 0) or {2,3} (half 1).
    const float4 fa = sC[i0 + col];
    v2f A;
    if (half == 0) { A[0] = -0.5f * fa.x; A[1] = -0.5f * fa.y; }  // (x_i, y_i)
    else           { A[0] = -0.5f * fa.z; A[1] = 1.0f; }          // (z_i, 1)

    // C accumulator: |c_i|^2 per row (VGPR v -> row v + 8*half).
    v8f Cacc;
    #pragma unroll
    for (int v = 0; v < 8; ++v) Cacc[v] = sC[i0 + v + 8 * half].w;

    float rmin[8];
    unsigned cntLo[8], cntHi[8];   // types 0/1 packed 16b in Lo, types 2/3 in Hi
    #pragma unroll
    for (int v = 0; v < 8; ++v) { rmin[v] = __builtin_inff(); cntLo[v] = 0u; cntHi[v] = 0u; }

    // B operand for tile jt: the two floats each half needs are contiguous
    // in the staged float4 (xy at +0, zw at +8) -> single ds_load_b64.
    auto tileD = [&](int jt, int j) -> v8f {
        const float2 fb = sC2[2 * j + half];
        v2f B; B[0] = fb.x; B[1] = fb.y;
        return __builtin_amdgcn_wmma_f32_16x16x4_f32(false, A, false, B,
                                                     (short)0, Cacc, false, false);
    };

    auto fastTile = [&](int jt) {
        const int j = (jt << 4) + col;
        const int t = sT[j];
        const v8f D = tileD(jt, j);
        const unsigned sh    = (unsigned)(t & 1) << 4;
        const unsigned incLo = (t & 2) ? 0u : (1u << sh);
        const unsigned incHi = (t & 2) ? (1u << sh) : 0u;
        #pragma unroll
        for (int v = 0; v < 8; ++v) {
            const float r = D[v];
            rmin[v] = (r < rmin[v]) ? r : rmin[v];
            const unsigned m = (r < RCUT2) ? 1u : 0u;
            cntLo[v] += m * incLo;
            cntHi[v] += m * incHi;
        }
    };

    const int jTiles = nloc >> 4;
    for (int jt = 0; jt < iTile; ++jt)          fastTile(jt);
    for (int jt = iTile + 1; jt < jTiles; ++jt) fastTile(jt);

    // Self tile: the only tile containing i==j pairs (row v+8*half == col).
    {
        const int j = (iTile << 4) + col;
        const int t = sT[j];
        const v8f D = tileD(iTile, j);
        const unsigned sh    = (unsigned)(t & 1) << 4;
        const unsigned incLo = (t & 2) ? 0u : (1u << sh);
        const unsigned incHi = (t & 2) ? (1u << sh) : 0u;
        #pragma unroll
        for (int v = 0; v < 8; ++v) {
            const bool self = (col == v + 8 * half);
            const float r = D[v];
            const unsigned m = (!self && (r < RCUT2)) ? 1u : 0u;
            rmin[v] = (!self && (r < rmin[v])) ? r : rmin[v];
            cntLo[v] += m * incLo;
            cntHi[v] += m * incHi;
        }
    }

    // Reduce across the 16 lanes of each half (wave32: width-16 xor shuffles).
    #pragma unroll
    for (int v = 0; v < 8; ++v) {
        float r = rmin[v];
        unsigned lo = cntLo[v], hi = cntHi[v];
        #pragma unroll
        for (int s = 1; s < 16; s <<= 1) {
            const float ro = __shfl_xor(r, s, 16);
            r = (ro < r) ? ro : r;
            lo += (unsigned)__shfl_xor((int)lo, s, 16);
            hi += (unsigned)__shfl_xor((int)hi, s, 16);
        }
        rmin[v] = r; cntLo[v] = lo; cntHi[v] = hi;
    }

    if (col == 0) {  // lanes 0 (rows 0-7) and 16 (rows 8-15)
        float* mo = minOut + (size_t)frame * nloc + i0 + 8 * half;
        int* wm = wsMax + frame * NTYPES_K;
        #pragma unroll
        for (int v = 0; v < 8; ++v) {
            mo[v] = rmin[v];
            atomicMax(&wm[0], (int)(cntLo[v] & 0xFFFFu));
            atomicMax(&wm[1], (int)(cntLo[v] >> 16));
            atomicMax(&wm[2], (int)(cntHi[v] & 0xFFFFu));
            atomicMax(&wm[3], (int)(cntHi[v] >> 16));
        }
    }
}

extern "C" void kernel_launch(void* const* d_in, const int* in_sizes, int n_in,
                              void* d_out, int out_size, void* d_ws, size_t ws_size,
                              hipStream_t stream) {
    const float* coord = (const float*)d_in[0];
    const int*   atype = (const int*)d_in[1];
    float* out   = (float*)d_out;
    int*   wsMax = (int*)d_ws;

    const int nA      = in_sizes[1];                    // nframes * nloc
    const int nframes = (out_size - nA) / NTYPES_K;     // out = nA + nframes*NTYPES
    const int nloc    = nA / nframes;

    ns_init_ws<<<1, 64, 0, stream>>>(wsMax, nframes * NTYPES_K);

    dim3 grid(nloc / 32, nframes);   // 2 i-tiles of 16 per block (2 waves)
    const size_t shmem = (size_t)nloc * (sizeof(float4) + sizeof(int));
    ns_main<<<grid, 64, shmem, stream>>>(coord, atype, out, wsMax, nloc);

    ns_finalize<<<1, 64, 0, stream>>>(wsMax, out + (size_t)nframes * nloc,
                                      nframes * NTYPES_K);
}